// BiGaBP_unfolding_lastiter_84731114815617
// MI455X (gfx1250) — compile-verified
//
#include <hip/hip_runtime.h>
#include <math.h>

#define NRr 16
#define NTt 8
#define Kk  64
#define PHI_ 1.0f
#define SQ2_ 1.41421356237309504880f
#define RS2_ 0.70710678118654752440f

#if defined(__HIP_DEVICE_COMPILE__) && defined(__gfx1250__) && \
    __has_builtin(__builtin_amdgcn_global_load_async_to_lds_b128) && \
    __has_builtin(__builtin_amdgcn_s_wait_asynccnt)
#define USE_ASYNC 1
#else
#define USE_ASYNC 0
#endif

typedef int v4i_t __attribute__((ext_vector_type(4)));
typedef __attribute__((address_space(1))) void  GASV;
typedef __attribute__((address_space(3))) void  LASV;
typedef __attribute__((address_space(1))) v4i_t GV4;
typedef __attribute__((address_space(3))) v4i_t LV4;

__device__ __forceinline__ void copy16(const void* g, void* l) {
#if USE_ASYNC
  __builtin_amdgcn_global_load_async_to_lds_b128((GV4*)(GASV*)g, (LV4*)(LASV*)l, 0, 0);
#else
  *(float4*)l = *(const float4*)g;
#endif
}
__device__ __forceinline__ void wait_slab_mid() {
#if USE_ASYNC
  __builtin_amdgcn_s_wait_asynccnt(12);   // 12 newer ops outstanding -> previous slab done
#endif
}
__device__ __forceinline__ void wait_slab_last() {
#if USE_ASYNC
  __builtin_amdgcn_s_wait_asynccnt(0);
#endif
}

struct __align__(16) SMem {
  float  Hs[2][NTt * Kk * 2];   // 2 x 4096 B
  float  Xs[2][NTt * Kk * 2];   // 2 x 4096 B
  float  vXs[2][NTt * Kk];      // 2 x 2048 B
  float  vHs[2][NTt * Kk];      // 2 x 2048 B
  float2 pn[8][NRr][NTt];       // 8-lane-group K-partials (num.re, num.im)  8 KB
  float  pd[8][NRr][NTt];       // den partials                              4 KB
};

__device__ __forceinline__ void issue_slab(const float* Hg, const float* Xg,
                                           const float* vXg, const float* vHg,
                                           SMem& sm, int b, int r2, int nb, int tid) {
  const size_t rowc = (size_t)(b * NRr + r2) * (NTt * Kk);     // complex offset of slab
  const char* hsrc = (const char*)(Hg + rowc * 2);
  const char* xsrc = (const char*)(Xg + rowc * 2);
  const char* vxs  = (const char*)(vXg + rowc);
  const char* vhs  = (const char*)(vHg + rowc);
  char* hdst = (char*)&sm.Hs[nb][0];
  char* xdst = (char*)&sm.Xs[nb][0];
  char* vxd  = (char*)&sm.vXs[nb][0];
  char* vhd  = (char*)&sm.vHs[nb][0];
  const int o = tid * 16;
#pragma unroll
  for (int i = 0; i < 4; ++i) copy16(hsrc + o + i * 1024, hdst + o + i * 1024);
#pragma unroll
  for (int i = 0; i < 4; ++i) copy16(xsrc + o + i * 1024, xdst + o + i * 1024);
#pragma unroll
  for (int i = 0; i < 2; ++i) copy16(vxs + o + i * 1024, vxd + o + i * 1024);
#pragma unroll
  for (int i = 0; i < 2; ++i) copy16(vhs + o + i * 1024, vhd + o + i * 1024);
}

__global__ __launch_bounds__(64)
void bigabp_last_iter_kernel(const float* __restrict__ Hg,  const float* __restrict__ Xg,
                             const float* __restrict__ vXg, const float* __restrict__ vHg,
                             const float* __restrict__ Yg,  const float* __restrict__ sN0,
                             const float* __restrict__ sAl, const float* __restrict__ sBe,
                             const float* __restrict__ sGa, const float* __restrict__ pmg,
                             float* __restrict__ out, int Bn) {
  __shared__ SMem sm;
  const int tid = threadIdx.x;          // tid == k
  const int k   = tid;
  const int b   = blockIdx.x;
  const int grp = tid >> 3;             // 8-lane group id, 0..7 across both waves

  const float N0    = sN0[0];
  const float alpha = sAl[0];
  const float beta  = sBe[0];
  const float gamma = sGa[0];
  const float pm    = pmg[(size_t)b * Kk + k];
  const float maskk = alpha * (1.0f - pm) + beta * pm;

  float accXr[NTt], accXi[NTt], accXd[NTt];
#pragma unroll
  for (int t = 0; t < NTt; ++t) { accXr[t] = 0.f; accXi[t] = 0.f; accXd[t] = 0.f; }

  issue_slab(Hg, Xg, vXg, vHg, sm, b, 0, 0, tid);

  for (int r = 0; r < NRr; ++r) {
    const int cur = r & 1;
    if (r + 1 < NRr) { issue_slab(Hg, Xg, vXg, vHg, sm, b, r + 1, cur ^ 1, tid); wait_slab_mid(); }
    else             { wait_slab_last(); }
    __syncthreads();                                   // slab r visible to both waves

    const float2 yv = ((const float2*)Yg)[(size_t)(b * NRr + r) * Kk + k];

    float hr[NTt], hi[NTt], xr[NTt], xi[NTt], vx[NTt], vh[NTt];
    float hxr[NTt], hxi[NTt], tm[NTt];
    float shr = 0.f, shi = 0.f, stm = 0.f;
#pragma unroll
    for (int t = 0; t < NTt; ++t) {
      const float2 h = *(const float2*)&sm.Hs[cur][(t * Kk + k) * 2];
      const float2 x = *(const float2*)&sm.Xs[cur][(t * Kk + k) * 2];
      const float vxx = sm.vXs[cur][t * Kk + k];
      const float vhh = sm.vHs[cur][t * Kk + k];
      hr[t] = h.x; hi[t] = h.y; xr[t] = x.x; xi[t] = x.y; vx[t] = vxx; vh[t] = vhh;
      hxr[t] = h.x * x.x - h.y * x.y;
      hxi[t] = h.x * x.y + h.y * x.x;
      shr += hxr[t]; shi += hxi[t];
      const float aH2 = h.x * h.x + h.y * h.y;
      const float aX2 = x.x * x.x + x.y * x.y;
      tm[t] = aH2 * vxx + vhh * (aX2 + vxx);
      stm += tm[t];
    }
#pragma unroll
    for (int t = 0; t < NTt; ++t) {
      const float er  = yv.x - shr + hxr[t];           // leave-one-out error
      const float ei  = yv.y - shi + hxi[t];
      const float xiy = stm - tm[t] + N0;
      const float rx  = 1.0f / (xiy + vh[t]);          // 1/xi_x
      const float rh  = 1.0f / (xiy + PHI_ * vx[t]);   // 1/xi_h
      const float aH2 = hr[t] * hr[t] + hi[t] * hi[t];
      accXd[t] += aH2 * rx;                            // sum_r |H|^2/xi_x
      accXr[t] += (hr[t] * er + hi[t] * ei) * rx;      // sum_r conj(H)*err/xi_x
      accXi[t] += (hr[t] * ei - hi[t] * er) * rx;
      const float w   = maskk * rh;
      const float aX2 = xr[t] * xr[t] + xi[t] * xi[t];
      float hd  = w * aX2;                             // mask*|X|^2/xi_h
      float hnr = w * (xr[t] * er + xi[t] * ei);       // mask*conj(X)*err/xi_h
      float hni = w * (xr[t] * ei - xi[t] * er);
#pragma unroll
      for (int off = 1; off <= 4; off <<= 1) {         // 3-step butterfly: 8-lane groups
        hnr += __shfl_xor(hnr, off, 32);
        hni += __shfl_xor(hni, off, 32);
        hd  += __shfl_xor(hd,  off, 32);
      }
      if ((tid & 7) == 0) {                            // one lane per 8-group stores
        float2 p; p.x = hnr; p.y = hni;
        sm.pn[grp][r][t] = p;
        sm.pd[grp][r][t] = hd;
      }
    }
    __syncthreads();                                   // reads done before next overwrite
  }

  // ---- X update outputs: X_est_[B,NT,K] (complex), var_X_[B,NT,K] ----
  {
    const float invg  = SQ2_ / gamma;
    const size_t xbase  = (size_t)Bn * NRr * NTt * 2;                 // after H_est_
    const size_t vxbase = xbase + (size_t)Bn * NTt * Kk * 2;          // after X_est_
#pragma unroll
    for (int t = 0; t < NTt; ++t) {
      const float vmsg = 1.0f / accXd[t];
      const float exr  = accXr[t] * vmsg;
      const float exi  = accXi[t] * vmsg;
      const float er   = tanhf(exr * invg);
      const float ei   = tanhf(exi * invg);
      const float vpost = 1.0f - 0.5f * (er * er + ei * ei);
      const size_t idx = (size_t)(b * NTt + t) * Kk + k;
      float2 xo; xo.x = er * RS2_ * pm; xo.y = ei * RS2_ * pm;
      *(float2*)&out[xbase + 2 * idx] = xo;
      out[vxbase + idx] = vpost * pm;
    }
  }

  // ---- H update outputs: H_est_[B,NR,NT] (complex), var_H_[B,NR,NT] ----
  {
    const size_t vhbase = (size_t)Bn * (NRr * NTt * 2 + NTt * Kk * 2 + NTt * Kk);
    for (int idx = tid; idx < NRr * NTt; idx += 64) {
      const int r = idx >> 3, t = idx & 7;
      float dn = 0.f, nr2 = 0.f, ni2 = 0.f;
#pragma unroll
      for (int g = 0; g < 8; ++g) {                    // fold 8 group-partials
        const float2 p = sm.pn[g][r][t];
        nr2 += p.x; ni2 += p.y; dn += sm.pd[g][r][t];
      }
      const float v = 1.0f / dn;                       // var_h_msg
      const float g = PHI_ / (PHI_ + v);               // Gaussian prior combine
      const size_t o = (size_t)b * NRr * NTt + idx;
      float2 ho; ho.x = g * (nr2 * v); ho.y = g * (ni2 * v);
      *(float2*)&out[2 * o] = ho;
      out[vhbase + o] = g * v;
    }
  }
}

extern "C" void kernel_launch(void* const* d_in, const int* in_sizes, int n_in,
                              void* d_out, int out_size, void* d_ws, size_t ws_size,
                              hipStream_t stream) {
  const float* Hg  = (const float*)d_in[0];
  const float* Xg  = (const float*)d_in[1];
  const float* vXg = (const float*)d_in[2];
  const float* vHg = (const float*)d_in[3];
  const float* Yg  = (const float*)d_in[4];
  const float* N0  = (const float*)d_in[5];
  const float* al  = (const float*)d_in[6];
  const float* be  = (const float*)d_in[7];
  const float* ga  = (const float*)d_in[8];
  const float* pmg = (const float*)d_in[9];
  float* out = (float*)d_out;

  int Bn = in_sizes[9] / Kk;   // pilot_mask holds B*K floats -> B
  if (Bn < 1) Bn = 1;
  bigabp_last_iter_kernel<<<Bn, 64, 0, stream>>>(Hg, Xg, vXg, vHg, Yg, N0, al, be, ga,
                                                 pmg, out, Bn);
}